// RankingLoss_14843406975506
// MI455X (gfx1250) — compile-verified
//
#include <hip/hip_runtime.h>
#include <hip/hip_bf16.h>

typedef __attribute__((ext_vector_type(2))) float v2f;
typedef __attribute__((ext_vector_type(8))) float v8f;

namespace {
constexpr int kB = 16384;          // batch rows (power of two)
constexpr int kD = 512;            // feature dim
constexpr int kL = 14;             // label dim
constexpr int kRowsPerTile = 15;   // outputs per 16x16 Gram tile (1-row overlap)
constexpr int kTiles = (kB + kRowsPerTile - 1) / kRowsPerTile;  // 1093
constexpr int kWavesPerWG = 4;
}

// One wave32 per tile: Gram tile C = Z_text[rows] * Z_image[rows]^T via
// V_WMMA_F32_16X16X4_F32, K-loop over D in steps of 4.
__global__ __launch_bounds__(kWavesPerWG * 32)
void gram_loss_kernel(const float* __restrict__ z_image,
                      const float* __restrict__ z_text,
                      const int*   __restrict__ labels,
                      float*       __restrict__ partial)
{
  __shared__ float gram[kWavesPerWG][16 * 16];

  const int lane = threadIdx.x & 31;
  const int wave = threadIdx.x >> 5;
  const int tile = blockIdx.x * kWavesPerWG + wave;
  if (tile >= kTiles) return;            // uniform per-wave exit; EXEC stays full

  const int i0 = tile * kRowsPerTile;
  const int lo = lane & 15;              // M index for A, N index for B
  const int hi = lane >> 4;              // selects K pair within each K=4 step

  // ISA 32-bit layouts: A 16x4 -> lane holds row (lane&15), cols k0+2*hi,+1
  //                     B 4x16 -> lane holds col (lane&15), rows k0+2*hi,+1
  // Identical per-lane float2 addressing for both operands.
  const int row = (i0 + lo) & (kB - 1);  // wrap only matters on the last tile
  const float* tp = z_text  + (size_t)row * kD + 2 * hi;  // A source (text rows)
  const float* ip = z_image + (size_t)row * kD + 2 * hi;  // B source (image cols)

  v8f acc = {0.f, 0.f, 0.f, 0.f, 0.f, 0.f, 0.f, 0.f};
#pragma unroll 4
  for (int k = 0; k < kD; k += 4) {
    v2f a = *(const v2f*)(tp + k);       // global_load_b64, 8B aligned
    v2f b = *(const v2f*)(ip + k);
    acc = __builtin_amdgcn_wmma_f32_16x16x4_f32(
        /*neg_a=*/false, a, /*neg_b=*/false, b,
        /*c_mod=*/(short)0, acc, /*reuse_a=*/false, /*reuse_b=*/false);
  }

  // Spill Gram tile to LDS: lane holds C[r + 8*hi][lo] in acc[r].
#pragma unroll
  for (int r = 0; r < 8; ++r)
    gram[wave][(r + 8 * hi) * 16 + lo] = acc[r];
  // DS ops from one wave execute in order; wait before cross-lane readback.
  asm volatile("s_wait_dscnt 0x0" ::: "memory");

  float li = 0.0f;
  const int i = lane;                    // lanes 0..14 own one output row each
  const int g = i0 + i;
  if (i < kRowsPerTile && g < kB) {
    const float paired  = gram[wave][i * 16 + i];            // <zi[g],  zt[g]>
    const float imp_img = gram[wave][i * 16 + (i + 1)];      // <zi[g+1],zt[g]>
    const float imp_txt = gram[wave][(i + 1) * 16 + i];      // <zt[g+1],zi[g]>

    const int j = (g + 1) & (kB - 1);
    const int* la = labels + (size_t)g * kL;
    const int* lb = labels + (size_t)j * kL;
    int un = 0, df = 0; bool eq = true;
#pragma unroll
    for (int t = 0; t < kL; ++t) {
      const int av = la[t], bv = lb[t];
      un += av | bv;
      df += av ^ bv;
      eq = eq && (av == bv);
    }
    const float margin = eq ? 0.0f
        : fmaxf(0.5f, (float)df / fmaxf((float)un, 1.0f));
    li = fmaxf(imp_img - paired + margin, 0.0f) +
         fmaxf(imp_txt - paired + margin, 0.0f);
  }

  // Deterministic wave32 butterfly reduction.
#pragma unroll
  for (int off = 16; off >= 1; off >>= 1)
    li += __shfl_xor(li, off, 32);
  if (lane == 0) partial[tile] = li;
}

__global__ __launch_bounds__(256)
void reduce_kernel(const float* __restrict__ partial, float* __restrict__ out)
{
  __shared__ float smem[256];
  float acc = 0.f;
  for (int t = threadIdx.x; t < kTiles; t += 256) acc += partial[t];
  smem[threadIdx.x] = acc;
  __syncthreads();
  for (int off = 128; off >= 1; off >>= 1) {
    if ((int)threadIdx.x < off) smem[threadIdx.x] += smem[threadIdx.x + off];
    __syncthreads();
  }
  if (threadIdx.x == 0) out[0] = smem[0] * (1.0f / (float)kB);
}

extern "C" void kernel_launch(void* const* d_in, const int* in_sizes, int n_in,
                              void* d_out, int out_size, void* d_ws, size_t ws_size,
                              hipStream_t stream)
{
  const float* z_image = (const float*)d_in[0];
  const float* z_text  = (const float*)d_in[1];
  const int*   labels  = (const int*)d_in[2];
  float* out     = (float*)d_out;
  float* partial = (float*)d_ws;   // kTiles floats (< 5 KB)

  const int grid = (kTiles + kWavesPerWG - 1) / kWavesPerWG;  // 274 WGs
  gram_loss_kernel<<<grid, kWavesPerWG * 32, 0, stream>>>(z_image, z_text,
                                                          labels, partial);
  reduce_kernel<<<1, 256, 0, stream>>>(partial, out);
}